// CrackLoss_70179765617291
// MI455X (gfx1250) — compile-verified
//
#include <hip/hip_runtime.h>
#include <math.h>

#define AS1 __attribute__((address_space(1)))
#define AS3 __attribute__((address_space(3)))

#if defined(__has_builtin)
#  if __has_builtin(__builtin_amdgcn_global_load_async_to_lds_b32)
#    define HAVE_ASYNC_LDS 1
#  endif
#  if __has_builtin(__builtin_amdgcn_s_wait_asynccnt)
#    define HAVE_WAIT_ASYNC 1
#  endif
#  if __has_builtin(__builtin_amdgcn_ballot_w32)
#    define HAVE_BALLOT32 1
#  endif
#endif
#ifndef HAVE_ASYNC_LDS
#define HAVE_ASYNC_LDS 0
#endif
#ifndef HAVE_WAIT_ASYNC
#define HAVE_WAIT_ASYNC 0
#endif
#ifndef HAVE_BALLOT32
#define HAVE_BALLOT32 0
#endif

// Problem constants (B,C,H,W) = (16,1,512,512)
constexpr int Bn = 16, Hn = 512, Wn = 512;
constexpr int TW = 256, TH = 32;          // tile (output) size
constexpr int GX = Wn / TW, GY = Hn / TH; // 2, 16
constexpr int NBLK = GX * GY * Bn;        // 512 blocks
constexpr int NQ = 11, QSTRIDE = 16;      // 11 partial sums, padded stride
constexpr int EW = TW + 2, EH = TH + 2;   // 258 x 34 logits tile (halo 1)
constexpr int WW = TW / 32 + 2;           // 10 packed words per row (word halo)
constexpr int WR = TH + 6;                // 38 rows in bit arrays (halo 3)
constexpr int OW = TW / 32;               // 8 output words per row

__device__ __forceinline__ float wave_reduce(float v) {
  for (int off = 16; off > 0; off >>= 1) v += __shfl_down(v, off, 32);
  return v;
}

__global__ __launch_bounds__(256) void crack_partials(
    const float* __restrict__ logits, const float* __restrict__ target,
    float* __restrict__ ws) {
  __shared__ float sD[EH][EW];          // logits tile, overwritten with (pred - t)
  __shared__ unsigned bitsD[WR][WW];    // packed target, OOB = 0 (dilation identity)
  __shared__ unsigned bitsE[WR][WW];    // packed target, OOB = 1 (erosion identity)
  __shared__ unsigned hd[3][WR][OW];    // horizontal dilation r=1,2,3
  __shared__ unsigned he[3][WR][OW];    // horizontal erosion  r=1,2,3
  __shared__ unsigned bnd[3][TH][OW];   // boundary masks k=3,5,7
  __shared__ float red[8 * NQ];

  const int tid = threadIdx.x;
  const int lane = tid & 31;
  const int wave = tid >> 5;
  const int bx = blockIdx.x, by = blockIdx.y, bz = blockIdx.z;
  const int tileX = bx * TW, tileY = by * TH;
  const size_t imgOff = (size_t)bz * Hn * Wn;
  const int tileWordX = tileX >> 5;

  // ---- Phase 0: stage logits tile (halo 1) into LDS via async copy ----
  for (int i = tid; i < EH * EW; i += 256) {
    int r = i / EW, c = i % EW;
    int y = tileY - 1 + r, x = tileX - 1 + c;
    if ((unsigned)y < (unsigned)Hn && (unsigned)x < (unsigned)Wn) {
      const float* g = logits + imgOff + (size_t)y * Wn + x;
      float* l = &sD[r][c];
#if HAVE_ASYNC_LDS
      __builtin_amdgcn_global_load_async_to_lds_b32((AS1 int*)g, (AS3 int*)l, 0, 0);
#else
      *l = *g;
#endif
    }
  }

  // ---- Phase 1: pack target bits via wave32 ballot (overlaps async copy) ----
  for (int w = wave; w < WR * WW; w += 8) {  // uniform per wave -> full EXEC for ballot
    int ri = w / WW, wi = w % WW;
    int y = tileY - 3 + ri;
    int x = ((tileWordX - 1 + wi) << 5) + lane;
    bool in = ((unsigned)y < (unsigned)Hn) && ((unsigned)x < (unsigned)Wn);
    float t = 0.f;
    if (in) t = target[imgOff + (size_t)y * Wn + x];
    bool bp = t > 0.5f;
#if HAVE_BALLOT32
    unsigned mD = __builtin_amdgcn_ballot_w32(in && bp);
    unsigned mE = __builtin_amdgcn_ballot_w32(!in || bp);
#else
    unsigned mD = (unsigned)__ballot(in && bp);
    unsigned mE = (unsigned)__ballot(!in || bp);
#endif
    if (lane == 0) { bitsD[ri][wi] = mD; bitsE[ri][wi] = mE; }
  }

#if HAVE_ASYNC_LDS
#  if HAVE_WAIT_ASYNC
  __builtin_amdgcn_s_wait_asynccnt(0);
#  else
  asm volatile("s_wait_asynccnt 0" ::: "memory");
#  endif
#endif
  __syncthreads();

  // ---- Phase 2: horizontal bitwise dilation/erosion, radii 1..3 ----
  for (int w = tid; w < WR * OW; w += 256) {
    int ri = w / OW, wo = w % OW, wi = wo + 1;
    unsigned C = bitsD[ri][wi], L = bitsD[ri][wi - 1], R = bitsD[ri][wi + 1];
    unsigned d = C;
    d |= (C << 1) | (L >> 31); d |= (C >> 1) | (R << 31); hd[0][ri][wo] = d;
    d |= (C << 2) | (L >> 30); d |= (C >> 2) | (R << 30); hd[1][ri][wo] = d;
    d |= (C << 3) | (L >> 29); d |= (C >> 3) | (R << 29); hd[2][ri][wo] = d;
    unsigned Ce = bitsE[ri][wi], Le = bitsE[ri][wi - 1], Re = bitsE[ri][wi + 1];
    unsigned e = Ce;
    e &= (Ce << 1) | (Le >> 31); e &= (Ce >> 1) | (Re << 31); he[0][ri][wo] = e;
    e &= (Ce << 2) | (Le >> 30); e &= (Ce >> 2) | (Re << 30); he[1][ri][wo] = e;
    e &= (Ce << 3) | (Le >> 29); e &= (Ce >> 3) | (Re << 29); he[2][ri][wo] = e;
  }
  __syncthreads();

  float a_bce = 0, a_pred = 0, a_tgt = 0, a_int = 0;
  float a_c3 = 0, a_b3 = 0, a_c5 = 0, a_b5 = 0, a_c7 = 0, a_b7 = 0, a_det = 0;

  // ---- Phase 3: vertical combine -> boundary masks, count via popc ----
  for (int w = tid; w < TH * OW; w += 256) {
    int ro = w / OW, wo = w % OW, ri = ro + 3;
    unsigned v3 = hd[0][ri - 1][wo] | hd[0][ri][wo] | hd[0][ri + 1][wo];
    unsigned e3 = he[0][ri - 1][wo] & he[0][ri][wo] & he[0][ri + 1][wo];
    unsigned v5 = hd[1][ri - 2][wo] | hd[1][ri - 1][wo] | hd[1][ri][wo] |
                  hd[1][ri + 1][wo] | hd[1][ri + 2][wo];
    unsigned e5 = he[1][ri - 2][wo] & he[1][ri - 1][wo] & he[1][ri][wo] &
                  he[1][ri + 1][wo] & he[1][ri + 2][wo];
    unsigned v7 = hd[2][ri - 3][wo] | hd[2][ri - 2][wo] | hd[2][ri - 1][wo] |
                  hd[2][ri][wo] | hd[2][ri + 1][wo] | hd[2][ri + 2][wo] |
                  hd[2][ri + 3][wo];
    unsigned e7 = he[2][ri - 3][wo] & he[2][ri - 2][wo] & he[2][ri - 1][wo] &
                  he[2][ri][wo] & he[2][ri + 1][wo] & he[2][ri + 2][wo] &
                  he[2][ri + 3][wo];
    unsigned m3 = v3 & ~e3, m5 = v5 & ~e5, m7 = v7 & ~e7;
    bnd[0][ro][wo] = m3; bnd[1][ro][wo] = m5; bnd[2][ro][wo] = m7;
    a_c3 += (float)__popc(m3);
    a_c5 += (float)__popc(m5);
    a_c7 += (float)__popc(m7);
  }
  __syncthreads();

  // ---- Phase 4: per-pixel sigmoid/BCE/sums; write D = pred - t in place ----
  for (int i = tid; i < EH * EW; i += 256) {
    int r = i / EW, c = i % EW;
    int y = tileY - 1 + r, x = tileX - 1 + c;
    bool in = ((unsigned)y < (unsigned)Hn) && ((unsigned)x < (unsigned)Wn);
    float dval = 0.f;
    if (in) {
      float xv = sD[r][c];
      int ri = r + 2;
      int wi = (x >> 5) - (tileWordX - 1);
      float t = (float)((bitsD[ri][wi] >> (x & 31)) & 1u);
      float p = 1.f / (1.f + __expf(-xv));
      dval = p - t;
      if (r >= 1 && r <= TH && c >= 1 && c <= TW) {  // interior (owned) pixel
        float bce = fmaxf(xv, 0.f) - xv * t + log1pf(__expf(-fabsf(xv)));
        a_bce += bce; a_pred += p; a_tgt += t; a_int += p * t;
        int ro = r - 1, co = c - 1, wo = co >> 5, bit = co & 31;
        a_b3 += bce * (float)((bnd[0][ro][wo] >> bit) & 1u);
        a_b5 += bce * (float)((bnd[1][ro][wo] >> bit) & 1u);
        a_b7 += bce * (float)((bnd[2][ro][wo] >> bit) & 1u);
      }
    }
    sD[r][c] = dval;  // 0 outside image => zero-padded conv input
  }
  __syncthreads();

  // ---- Phase 5: Laplacian |lap(pred - t)| from LDS ----
  for (int i = tid; i < TH * TW; i += 256) {
    int ro = i / TW, co = i % TW;
    int r = ro + 1, c = co + 1;
    float lap = sD[r - 1][c] + sD[r + 1][c] + sD[r][c - 1] + sD[r][c + 1] -
                4.f * sD[r][c];
    a_det += fabsf(lap);
  }

  // ---- Block reduction (deterministic), per-block partials to d_ws ----
  float acc[NQ] = {a_bce, a_pred, a_tgt, a_int, a_c3, a_b3,
                   a_c5,  a_b5,   a_c7,  a_b7,  a_det};
  for (int q = 0; q < NQ; ++q) {
    float v = wave_reduce(acc[q]);
    if (lane == 0) red[wave * NQ + q] = v;
  }
  __syncthreads();
  if (tid < NQ) {
    float s = 0.f;
    for (int w = 0; w < 8; ++w) s += red[w * NQ + tid];
    int blk = bx + GX * (by + GY * bz);
    ws[blk * QSTRIDE + tid] = s;
  }
}

__global__ __launch_bounds__(256) void crack_finalize(
    const float* __restrict__ ws, float* __restrict__ out) {
  __shared__ double red[256];
  __shared__ double tot[NQ];
  const int tid = threadIdx.x;
  for (int q = 0; q < NQ; ++q) {
    double s = 0.0;
    for (int b = tid; b < NBLK; b += 256) s += (double)ws[b * QSTRIDE + q];
    red[tid] = s;
    __syncthreads();
    for (int off = 128; off > 0; off >>= 1) {
      if (tid < off) red[tid] += red[tid + off];
      __syncthreads();
    }
    if (tid == 0) tot[q] = red[0];
    __syncthreads();
  }
  if (tid == 0) {
    const double N = (double)Bn * Hn * Wn;
    double bce = tot[0] / N;
    double sp = tot[1], st = tot[2], si = tot[3];
    double dice = 1.0 - (2.0 * si + 1.0) / (sp + st + 1.0);
    double fp = sp - si, fn = st - si;
    double ti = (si + 1.0) / (si + 0.6 * fp + 0.4 * fn + 1.0);
    double tversky = pow(fmax(1.0 - ti, 0.0), 0.75);
    double cs[3] = {tot[4], tot[6], tot[8]};
    double bs[3] = {tot[5], tot[7], tot[9]};
    double tb = 0.0, ns = 0.0;
    for (int k = 0; k < 3; ++k)
      if (cs[k] >= 1.0) { tb += bs[k] / fmax(cs[k], 1.0); ns += 1.0; }
    double boundary = tb / fmax(ns, 1.0);
    double detail = tot[10] / N;
    out[0] = (float)(bce + dice + 0.5 * tversky + 0.5 * boundary + 0.3 * detail);
  }
}

extern "C" void kernel_launch(void* const* d_in, const int* in_sizes, int n_in,
                              void* d_out, int out_size, void* d_ws,
                              size_t ws_size, hipStream_t stream) {
  (void)in_sizes; (void)n_in; (void)out_size; (void)ws_size;
  const float* logits = (const float*)d_in[0];
  const float* target = (const float*)d_in[1];
  float* ws = (float*)d_ws;  // needs NBLK*QSTRIDE*4 = 32 KiB
  dim3 grid(GX, GY, Bn), block(256);
  hipLaunchKernelGGL(crack_partials, grid, block, 0, stream, logits, target, ws);
  hipLaunchKernelGGL(crack_finalize, dim3(1), dim3(256), 0, stream, ws,
                     (float*)d_out);
}